// RGAT_MaxMargin_Stack_with_Attention_Info_68281390072292
// MI455X (gfx1250) — compile-verified
//
#include <hip/hip_runtime.h>
#include <hip/hip_bf16.h>

// Problem constants from the reference.
#define B_ 128
#define L_ 512
#define D_ 768
#define H_ 8
#define R_ 32

typedef float v2f __attribute__((ext_vector_type(2)));
typedef float v8f __attribute__((ext_vector_type(8)));

// Output layout in d_out (flat, f32): score[128] | loss[1] | head_attn[128*512] | tail_attn[128*512]
#define OUT_SCORE   0
#define OUT_LOSS    128
#define OUT_ATTN    129

// ---------------------------------------------------------------------------
// Kernel 1: attention pool (one workgroup per (batch, pool)).
//   - logits via V_WMMA_F32_16X16X4_F32 (fp32 WMMA; memory-bound problem so
//     full precision costs nothing)
//   - softmax(relu) per head (one wave32 per head)
//   - head-mean of attn (output) then GEMV pooling against value
// ---------------------------------------------------------------------------
__global__ __launch_bounds__(256) void attn_pool_kernel(
    const float* __restrict__ head_value,
    const float* __restrict__ tail_value,
    const float* __restrict__ prop_embed,
    const float* __restrict__ W_att,
    const float* __restrict__ b_att,
    const int*   __restrict__ head_ids,
    const int*   __restrict__ tail_ids,
    const int*   __restrict__ rc_ids,
    float*       __restrict__ d_out,
    float*       __restrict__ ws_cat)   // [B][2*D]
{
    const int b    = blockIdx.x;
    const int pool = blockIdx.y;
    const float* value = pool ? tail_value : head_value;
    const int*   ids   = pool ? tail_ids  : head_ids;
    float* attn_out = d_out + OUT_ATTN + (size_t)pool * (B_ * L_) + (size_t)b * L_;

    const int tid  = threadIdx.x;
    const int lane = tid & 31;
    const int wave = tid >> 5;

    // sW has 9 rows: rows 0..7 = Wsel, row 8 = zeros. Padded WMMA lanes
    // (M = 8..15) read the zero row, so the A fragment needs no VALU masking.
    __shared__ float sW[(H_ + 1) * D_];  // 9 x 768                    (27 KB)
    __shared__ float sA[H_ * L_];        // logits -> attn: 8 x 512    (16 KB)
    __shared__ float sMean[L_];          // head-averaged attn         (2 KB)
    __shared__ float sBias[H_];

    const int rc = rc_ids[b];

    // Stage Wsel[h,:] = W_att[h, rc, :] into LDS; zero the padding row.
    for (int i = tid; i < (H_ + 1) * D_; i += 256) {
        const int h = i / D_;
        const int d = i - h * D_;
        sW[i] = (h < H_) ? W_att[((size_t)h * R_ + rc) * D_ + d] : 0.0f;
    }
    if (tid < H_) sBias[tid] = b_att[tid * R_ + rc];
    __syncthreads();

    // ---- Phase 2: logits[h, l] = Wsel[h,:] . prop_embed[ids[b,l], :] -------
    // 32 tiles of 16 L-columns; 8 waves -> 4 tiles each.
    // A (16x4 f32): lanes 0-15 -> M=lane, K={k,k+1}; lanes 16-31 -> M=lane-16,
    //               K={k+2,k+3}.  Rows M>=8 read the LDS zero row.
    // B (4x16 f32): lane n -> column N=n&15, same K split as A.
    const int half = lane >> 4;          // 0: lanes 0-15, 1: lanes 16-31
    const int sub  = lane & 15;          // M (for A) / N (for B)
    const int koff = half * 2;
    const int wrow = (sub < H_) ? sub : H_;      // padded rows -> zero row
    const float* wr = sW + (size_t)wrow * D_ + koff;

    for (int t = wave; t < L_ / 16; t += 8) {
        const int l0  = t * 16;
        const int row = ids[(size_t)b * L_ + l0 + sub];
        const float* fr = prop_embed + (size_t)row * D_ + koff;

        v8f c = {};
        #pragma unroll 8
        for (int k = 0; k < D_; k += 4) {
            v2f a;
            a.x = wr[k];
            a.y = wr[k + 1];
            v2f bb;
            bb.x = fr[k];
            bb.y = fr[k + 1];
            c = __builtin_amdgcn_wmma_f32_16x16x4_f32(
                    /*neg_a=*/false, a, /*neg_b=*/false, bb,
                    /*c_mod=*/(short)0, c, /*reuse_a=*/false, /*reuse_b=*/false);
        }
        // C layout: VGPR r, lanes 0-15 -> M=r (head r), N=lane. Lanes 16-31
        // hold padded heads 8-15 -> discard.
        if (half == 0) {
            #pragma unroll
            for (int r = 0; r < H_; ++r)
                sA[r * L_ + l0 + sub] = c[r] + sBias[r];
        }
    }
    __syncthreads();

    // ---- Phase 3: attn[h,:] = softmax(relu(logits[h,:])) over L ------------
    {
        const int h = wave;              // 8 waves == 8 heads
        float vals[L_ / 32];
        float vmax = 0.0f;               // relu output >= 0
        #pragma unroll
        for (int i = 0; i < L_ / 32; ++i) {
            float x = sA[h * L_ + lane + i * 32];
            x = x > 0.0f ? x : 0.0f;
            vals[i] = x;
            vmax = fmaxf(vmax, x);
        }
        #pragma unroll
        for (int off = 16; off >= 1; off >>= 1)
            vmax = fmaxf(vmax, __shfl_xor(vmax, off, 32));
        float vsum = 0.0f;
        #pragma unroll
        for (int i = 0; i < L_ / 32; ++i) {
            const float e = __expf(vals[i] - vmax);
            vals[i] = e;
            vsum += e;
        }
        #pragma unroll
        for (int off = 16; off >= 1; off >>= 1)
            vsum += __shfl_xor(vsum, off, 32);
        const float inv = 1.0f / vsum;
        #pragma unroll
        for (int i = 0; i < L_ / 32; ++i)
            sA[h * L_ + lane + i * 32] = vals[i] * inv;
    }
    __syncthreads();

    // ---- Phase 4: attn_mean = mean over heads (output + pooling weights) ---
    for (int l = tid; l < L_; l += 256) {
        float s = 0.0f;
        #pragma unroll
        for (int h = 0; h < H_; ++h) s += sA[h * L_ + l];
        s *= (1.0f / H_);
        sMean[l]    = s;
        attn_out[l] = s;
    }
    __syncthreads();

    // ---- Phase 5: out[b,:] = attn_mean[b,:] @ value[b,:,:]  (GEMV) ---------
    // (einsum('hbl,bld->bd', attn, value)/H  ==  attn_mean @ value)
    const float* vb = value + (size_t)b * L_ * D_;
    const int d0 = tid, d1 = tid + 256, d2 = tid + 512;
    float acc0 = 0.0f, acc1 = 0.0f, acc2 = 0.0f;
    for (int l = 0; l < L_; ++l) {
        const float w = sMean[l];
        const float* vr = vb + (size_t)l * D_;
        acc0 += w * vr[d0];
        acc1 += w * vr[d1];
        acc2 += w * vr[d2];
    }
    float* cat = ws_cat + (size_t)b * (2 * D_) + (size_t)pool * D_;
    cat[d0] = acc0; cat[d1] = acc1; cat[d2] = acc2;
}

// ---------------------------------------------------------------------------
// Kernel 2: score[b] = cat[b,:] . W1[rc[b],:] + b1[rc[b]]
// ---------------------------------------------------------------------------
__global__ __launch_bounds__(256) void score_kernel(
    const float* __restrict__ ws_cat,
    const float* __restrict__ W1,
    const float* __restrict__ b1,
    const int*   __restrict__ rc_ids,
    float*       __restrict__ d_out)
{
    const int b   = blockIdx.x;
    const int tid = threadIdx.x;
    const int rc  = rc_ids[b];
    const float* cat = ws_cat + (size_t)b * (2 * D_);
    const float* w   = W1 + (size_t)rc * (2 * D_);

    float s = 0.0f;
    for (int i = tid; i < 2 * D_; i += 256) s += cat[i] * w[i];

    __shared__ float red[8];
    #pragma unroll
    for (int off = 16; off >= 1; off >>= 1) s += __shfl_xor(s, off, 32);
    if ((tid & 31) == 0) red[tid >> 5] = s;
    __syncthreads();
    if (tid == 0) {
        float t = 0.0f;
        #pragma unroll
        for (int i = 0; i < 8; ++i) t += red[i];
        d_out[OUT_SCORE + b] = t + b1[rc];
    }
}

// ---------------------------------------------------------------------------
// Kernel 3: loss = mean(relu(1 - (score[:64] - score[64:])))
// ---------------------------------------------------------------------------
__global__ __launch_bounds__(64) void loss_kernel(float* __restrict__ d_out)
{
    const int tid = threadIdx.x;   // 64 threads
    const float diff = 1.0f - (d_out[OUT_SCORE + tid] - d_out[OUT_SCORE + tid + 64]);
    float s = diff > 0.0f ? diff : 0.0f;

    __shared__ float red[2];
    #pragma unroll
    for (int off = 16; off >= 1; off >>= 1) s += __shfl_xor(s, off, 32);
    if ((tid & 31) == 0) red[tid >> 5] = s;
    __syncthreads();
    if (tid == 0) d_out[OUT_LOSS] = (red[0] + red[1]) * (1.0f / 64.0f);
}

// ---------------------------------------------------------------------------
extern "C" void kernel_launch(void* const* d_in, const int* in_sizes, int n_in,
                              void* d_out, int out_size, void* d_ws, size_t ws_size,
                              hipStream_t stream) {
    const float* head_value = (const float*)d_in[0];
    const float* tail_value = (const float*)d_in[1];
    const float* prop_embed = (const float*)d_in[2];
    const float* W_att      = (const float*)d_in[3];
    const float* b_att      = (const float*)d_in[4];
    const float* W1         = (const float*)d_in[5];
    const float* b1         = (const float*)d_in[6];
    const int*   head_ids   = (const int*)d_in[7];
    const int*   tail_ids   = (const int*)d_in[8];
    const int*   rc_ids     = (const int*)d_in[9];
    // d_in[10] (nd_labels) unused by the reference computation.

    float* out    = (float*)d_out;
    float* ws_cat = (float*)d_ws;   // B * 2D floats = 768 KiB

    dim3 grid(B_, 2);               // (batch, pool)
    attn_pool_kernel<<<grid, 256, 0, stream>>>(head_value, tail_value, prop_embed,
                                               W_att, b_att, head_ids, tail_ids,
                                               rc_ids, out, ws_cat);
    score_kernel<<<B_, 256, 0, stream>>>(ws_cat, W1, b1, rc_ids, out);
    loss_kernel<<<1, 64, 0, stream>>>(out);
}